// Features_4D_To_2D_2362232013156
// MI455X (gfx1250) — compile-verified
//
#include <hip/hip_runtime.h>
#include <stdint.h>

// im2col: in (16,112,112,64) f32 -> out (16, 110*110, 3*3*64) f32
// Store-BW bound (~446 MB out, ~21 us floor @ 23.3 TB/s). Data path:
//   global --TDM tensor_load_to_lds--> LDS (9x reuse) --async-from-LDS--> global (NT)
// One TDM descriptor per workgroup fills the 12x12x64 tile; waves only do
// strength-reduced address math for the output store stream.

typedef float    v4f  __attribute__((ext_vector_type(4)));
typedef uint32_t u32x4 __attribute__((ext_vector_type(4)));
typedef uint32_t u32x8 __attribute__((ext_vector_type(8)));

#define KHH   3
#define KWW   3
#define TILE  10                        // output positions per dim per workgroup
#define IN_W  112
#define IN_H  112
#define CH    64
#define NH    110
#define NV    110
#define C4    (CH / 4)                  // 16 float4 per pixel
#define TIN   (TILE + KHH - 1)          // 12 input rows/cols per tile
#define TILE_V4 (TIN * TIN * C4)        // 2304 float4 staged in LDS (36 KB)
#define ROW_V4  (KHH * KWW * C4)        // 144 float4 per output row
#define NT_THREADS 288                  // 9 waves; 288 = 2 * 144

#define TLINE   (TIN * CH)              // 768 contiguous floats per tile line
#define LSTRIDE (IN_H * CH)             // 7168 floats between input lines

__launch_bounds__(NT_THREADS, 2)
__global__ void im2col_tdm_kernel(const float* __restrict__ in,
                                  float* __restrict__ out)
{
    __shared__ v4f tile[TILE_V4];

    const int b   = blockIdx.z;
    const int h0  = blockIdx.y * TILE;
    const int v0  = blockIdx.x * TILE;
    const int tid = threadIdx.x;

    // ---------- Phase 1: TDM fill of the 12x12x64 tile (one op per WG) ----------
    // 2D tensor: 12 lines x 768 contiguous f32, line stride 7168 f32, linear in LDS.
    if (tid < 32) {                      // wave-uniform: only wave 0 issues the TDM op
        const uint64_t ga  = (uint64_t)(uintptr_t)
            (in + (((size_t)b * IN_W + h0) * IN_H + v0) * CH);
        const uint32_t lds = (uint32_t)(uintptr_t)tile;    // WG-relative LDS offset

        u32x4 g0;                                          // D# group 0 (128b)
        g0.x = 1u;                                         // count=1, user mode
        g0.y = lds;                                        // lds_addr [63:32]
        g0.z = (uint32_t)ga;                               // global_addr[31:0]
        g0.w = (uint32_t)(ga >> 32) | (2u << 30);          // global_addr[56:32] | type=2

        u32x8 g1;                                          // D# group 1 (256b)
        g1.s0 = 2u << 16;                                  // data_size=4B; mask=0; no pad
        g1.s1 = (uint32_t)(TLINE & 0xFFFFu) << 16;         // tensor_dim0[15:0] @ [63:48]
        g1.s2 = (uint32_t)(TLINE >> 16) | ((uint32_t)TIN << 16); // dim0 hi | tensor_dim1=12
        g1.s3 = ((uint32_t)TLINE << 16);                   // tile_dim0=768 @ [127:112]
        g1.s4 = (uint32_t)TIN;                             // tile_dim1=12, tile_dim2=0
        g1.s5 = (uint32_t)LSTRIDE;                         // tensor_dim0_stride[31:0]=7168
        g1.s6 = 0u;                                        // stride hi, dim1_stride lo
        g1.s7 = 0u;                                        // dim1_stride hi

        asm volatile("tensor_load_to_lds %0, %1" :: "s"(g0), "s"(g1) : "memory");
        __builtin_amdgcn_s_wait_tensorcnt(0);
    }
    __syncthreads();

    // ---------- Phase 2: LDS -> global via async-from-LDS, NT b128 stores ----------
    // Thread owns fixed column q of the 144-float4 output row; walks pos=rid,rid+2,...
    // out[b, (h0+dh)*110+(v0+dv), q*4..] = tile[((dh+kh)*12 + dv+kw)*16 + d4]
    const int q   = tid % ROW_V4;        // 0..143  (fixed per thread)
    const int rid = tid / ROW_V4;        // 0..1
    const int k   = q >> 4;              // 0..8  (kh*3+kw)
    const int d4  = q & 15;              // 0..15
    const int kh  = k / KWW;
    const int kw  = k - kh * KWW;

    uint32_t ldsoff = (uint32_t)(uintptr_t)(&tile[(kh * TIN + kw) * C4 + d4])
                    + (uint32_t)rid * (C4 * 16);                  // dv = rid
    const size_t orow0 = (size_t)b * (NH * NV) + (size_t)h0 * NV + v0;
    uint64_t gout = (uint64_t)(uintptr_t)out
                  + (orow0 * ROW_V4 + (size_t)q) * 16
                  + (uint64_t)rid * (ROW_V4 * 16);                // dv = rid
    int dv = rid;

    #pragma unroll 5
    for (int it = 0; it < (TILE * TILE) / 2; ++it) {              // 50 positions/thread
        asm volatile("global_store_async_from_lds_b128 %0, %1, off th:TH_STORE_NT"
                     :: "v"(gout), "v"(ldsoff) : "memory");
        int  dvn  = dv + 2;
        bool wrap = dvn >= TILE;
        dv = wrap ? dvn - TILE : dvn;
        ldsoff += wrap ? 4u * (C4 * 16) : 2u * (C4 * 16);         // +1024 / +512 B
        gout   += wrap ? (uint64_t)(NV - TILE + 2) * (ROW_V4 * 16)
                       : (uint64_t)2 * (ROW_V4 * 16);             // +235008 / +4608 B
    }
    asm volatile("s_wait_asynccnt 0" ::: "memory");               // drain before endpgm
}

extern "C" void kernel_launch(void* const* d_in, const int* in_sizes, int n_in,
                              void* d_out, int out_size, void* d_ws, size_t ws_size,
                              hipStream_t stream)
{
    const float* in  = (const float*)d_in[0];
    float*       out = (float*)d_out;
    const int B = in_sizes[0] / (IN_W * IN_H * CH);   // 16
    dim3 grid(NV / TILE, NH / TILE, B);               // (11, 11, 16)
    im2col_tdm_kernel<<<grid, NT_THREADS, 0, stream>>>(in, out);
}